// DirGATConv_74861279969845
// MI455X (gfx1250) — compile-verified
//
#include <hip/hip_runtime.h>
#include <math.h>

#define NNODES 50000
#define NEDGES 800000
#define DIN    128
#define DOUT   64
#define ALPHA_MIX 0.5f
#define NEG_SLOPE 0.2f
#define ROWBLOCKS (NNODES / 16)   // 3125, exact

typedef __attribute__((ext_vector_type(2))) float v2f;
typedef __attribute__((ext_vector_type(8))) float v8f;

// ---------------------------------------------------------------------------
// Init: zero output accumulator, -inf for segment-max, 0 for denominators.
// ---------------------------------------------------------------------------
__global__ void dirgat_init(float* __restrict__ out,
                            float* __restrict__ m1, float* __restrict__ m2,
                            float* __restrict__ den1, float* __restrict__ den2) {
  int t = blockIdx.x * blockDim.x + threadIdx.x;
  if (t < NNODES * DOUT) out[t] = 0.0f;
  if (t < NNODES) {
    m1[t] = -INFINITY; m2[t] = -INFINITY;
    den1[t] = 0.0f;    den2[t] = 0.0f;
  }
}

// ---------------------------------------------------------------------------
// xw = x @ W via V_WMMA_F32_16X16X4_F32 (fp32 in, fp32 accumulate).
// One wave computes a 16-row x 64-col tile of xw (4 WMMA column tiles),
// marching K=128 in steps of 4. blockIdx.y selects W1/W2 direction.
//
// Fragment layouts (CDNA5 ISA 7.12.2):
//   A 16x4 f32: lane%16 = row M, VGPR0/1 = K pair, lane/16 selects K={0,1} vs {2,3}
//   B 4x16 f32: lane%16 = col N, same K striping as A
//   C/D 16x16 f32: VGPR r -> row r + 8*(lane/16), lane%16 = col
// ---------------------------------------------------------------------------
__global__ void dirgat_gemm_wmma(const float* __restrict__ x,
                                 const float* __restrict__ W1,
                                 const float* __restrict__ W2,
                                 float* __restrict__ xw1,
                                 float* __restrict__ xw2) {
  const float* W  = blockIdx.y ? W2  : W1;
  float*       xw = blockIdx.y ? xw2 : xw1;

  const int wave = threadIdx.x >> 5;
  const int lane = threadIdx.x & 31;
  const int rb   = blockIdx.x * (blockDim.x >> 5) + wave;
  if (rb >= ROWBLOCKS) return;            // wave-uniform exit (EXEC all-1 for WMMA)

  const int m     = lane & 15;            // row (A) / col (B,C) within tile
  const int khalf = (lane >> 4) * 2;      // which K-pair this half-wave owns

  const float* xrow = x + (size_t)(rb * 16 + m) * DIN;

  v8f acc0 = 0.0f, acc1 = 0.0f, acc2 = 0.0f, acc3 = 0.0f;

#pragma unroll 4
  for (int kb = 0; kb < DIN / 4; ++kb) {
    const int kk = kb * 4 + khalf;
    v2f a;
    a.x = xrow[kk];
    a.y = xrow[kk + 1];

    const float* Wr0 = W + (size_t)kk * DOUT + m;        // row kk   of W
    const float* Wr1 = Wr0 + DOUT;                       // row kk+1 of W

    v2f b;
    b.x = Wr0[0];  b.y = Wr1[0];
    acc0 = __builtin_amdgcn_wmma_f32_16x16x4_f32(false, a, false, b, (short)0, acc0, false, false);
    b.x = Wr0[16]; b.y = Wr1[16];
    acc1 = __builtin_amdgcn_wmma_f32_16x16x4_f32(false, a, false, b, (short)0, acc1, false, false);
    b.x = Wr0[32]; b.y = Wr1[32];
    acc2 = __builtin_amdgcn_wmma_f32_16x16x4_f32(false, a, false, b, (short)0, acc2, false, false);
    b.x = Wr0[48]; b.y = Wr1[48];
    acc3 = __builtin_amdgcn_wmma_f32_16x16x4_f32(false, a, false, b, (short)0, acc3, false, false);
  }

  // Store C/D tiles: VGPR r holds row (r + 8*(lane>>4)), column m of each tile.
  const int rbase = rb * 16 + (lane >> 4) * 8;
#pragma unroll
  for (int r = 0; r < 8; ++r) {
    float* o = xw + (size_t)(rbase + r) * DOUT + m;
    o[0]  = acc0[r];
    o[16] = acc1[r];
    o[32] = acc2[r];
    o[48] = acc3[r];
  }
}

// ---------------------------------------------------------------------------
// Per-node attention scores: a_s = xw . att_src ; a_d = xw . att_dst
// ---------------------------------------------------------------------------
__global__ void dirgat_scores(const float* __restrict__ xw1, const float* __restrict__ xw2,
                              const float* __restrict__ as1w, const float* __restrict__ ad1w,
                              const float* __restrict__ as2w, const float* __restrict__ ad2w,
                              float* __restrict__ a_s1, float* __restrict__ a_d1,
                              float* __restrict__ a_s2, float* __restrict__ a_d2) {
  int n = blockIdx.x * blockDim.x + threadIdx.x;
  if (n >= NNODES) return;
  const float* r1 = xw1 + (size_t)n * DOUT;
  const float* r2 = xw2 + (size_t)n * DOUT;
  float s1 = 0.f, d1 = 0.f, s2 = 0.f, d2 = 0.f;
#pragma unroll 8
  for (int c = 0; c < DOUT; ++c) {
    float v1 = r1[c], v2 = r2[c];
    s1 += v1 * as1w[c]; d1 += v1 * ad1w[c];
    s2 += v2 * as2w[c]; d2 += v2 * ad2w[c];
  }
  a_s1[n] = s1; a_d1[n] = d1;
  a_s2[n] = s2; a_d2[n] = d2;
}

__device__ __forceinline__ float leakyr(float v) {
  return v > 0.0f ? v : NEG_SLOPE * v;
}

__device__ __forceinline__ void atomicMaxF(float* addr, float val) {
  int* ai = (int*)addr;
  int old = __hip_atomic_load(ai, __ATOMIC_RELAXED, __HIP_MEMORY_SCOPE_AGENT);
  while (__int_as_float(old) < val) {
    int assumed = old;
    old = atomicCAS(ai, assumed, __float_as_int(val));
    if (old == assumed) break;
  }
}

// ---------------------------------------------------------------------------
// Edge pass 1: segment max of leaky-relu scores per destination.
// Virtual edges i >= NEDGES are the self-loops (s == d == i - NEDGES).
// Direction 2 uses transposed edges: src2 = dst, dst2 = src.
// ---------------------------------------------------------------------------
__global__ void dirgat_edge_max(const int* __restrict__ src, const int* __restrict__ dst,
                                const float* __restrict__ a_s1, const float* __restrict__ a_d1,
                                const float* __restrict__ a_s2, const float* __restrict__ a_d2,
                                float* __restrict__ m1, float* __restrict__ m2) {
  int i = blockIdx.x * blockDim.x + threadIdx.x;
  if (i >= NEDGES + NNODES) return;
  int s, d;
  if (i < NEDGES) { s = src[i]; d = dst[i]; } else { s = d = i - NEDGES; }
  float e1 = leakyr(a_s1[s] + a_d1[d]);
  atomicMaxF(&m1[d], e1);
  float e2 = leakyr(a_s2[d] + a_d2[s]);   // reversed direction: dst2 = s
  atomicMaxF(&m2[s], e2);
}

// ---------------------------------------------------------------------------
// Edge pass 2: ex = exp(e - m[dst]); denom[dst] += ex. Stores ex per edge.
// ---------------------------------------------------------------------------
__global__ void dirgat_edge_exp(const int* __restrict__ src, const int* __restrict__ dst,
                                const float* __restrict__ a_s1, const float* __restrict__ a_d1,
                                const float* __restrict__ a_s2, const float* __restrict__ a_d2,
                                const float* __restrict__ m1, const float* __restrict__ m2,
                                float* __restrict__ den1, float* __restrict__ den2,
                                float* __restrict__ ex1, float* __restrict__ ex2) {
  int i = blockIdx.x * blockDim.x + threadIdx.x;
  if (i >= NEDGES + NNODES) return;
  int s, d;
  if (i < NEDGES) { s = src[i]; d = dst[i]; } else { s = d = i - NEDGES; }

  float mm1 = m1[d]; if (mm1 < -3.0e38f) mm1 = 0.0f;  // isfinite guard
  float v1 = expf(leakyr(a_s1[s] + a_d1[d]) - mm1);
  ex1[i] = v1;
  atomicAdd(&den1[d], v1);

  float mm2 = m2[s]; if (mm2 < -3.0e38f) mm2 = 0.0f;
  float v2 = expf(leakyr(a_s2[d] + a_d2[s]) - mm2);
  ex2[i] = v2;
  atomicAdd(&den2[s], v2);
}

// ---------------------------------------------------------------------------
// Edge pass 3: normalize into softmax coefficients (in place on ex buffers).
// ---------------------------------------------------------------------------
__global__ void dirgat_edge_coef(const int* __restrict__ src, const int* __restrict__ dst,
                                 const float* __restrict__ den1, const float* __restrict__ den2,
                                 float* __restrict__ ex1, float* __restrict__ ex2) {
  int i = blockIdx.x * blockDim.x + threadIdx.x;
  if (i >= NEDGES + NNODES) return;
  int s, d;
  if (i < NEDGES) { s = src[i]; d = dst[i]; } else { s = d = i - NEDGES; }
  ex1[i] = ex1[i] / (den1[d] + 1e-16f);
  ex2[i] = ex2[i] / (den2[s] + 1e-16f);
}

// ---------------------------------------------------------------------------
// Edge pass 4: scatter-add. One thread per (edge, feature). The (1-a)/a mixing
// is folded into the scatter so both directions accumulate into d_out.
// ---------------------------------------------------------------------------
__global__ void dirgat_edge_scatter(const int* __restrict__ src, const int* __restrict__ dst,
                                    const float* __restrict__ coef1, const float* __restrict__ coef2,
                                    const float* __restrict__ xw1, const float* __restrict__ xw2,
                                    float* __restrict__ out) {
  long long t = (long long)blockIdx.x * blockDim.x + threadIdx.x;
  int e = (int)(t >> 6);
  int c = (int)(t & 63);
  if (e >= NEDGES + NNODES) return;
  int s, d;
  if (e < NEDGES) { s = src[e]; d = dst[e]; } else { s = d = e - NEDGES; }

  // forward: out[d] += (1-a) * coef1 * xw1[s]
  atomicAdd(&out[(size_t)d * DOUT + c],
            (1.0f - ALPHA_MIX) * coef1[e] * xw1[(size_t)s * DOUT + c]);
  // reverse: out[s] += a * coef2 * xw2[d]   (dst2 = s, src2 = d)
  atomicAdd(&out[(size_t)s * DOUT + c],
            ALPHA_MIX * coef2[e] * xw2[(size_t)d * DOUT + c]);
}

// ---------------------------------------------------------------------------
// Final: blended bias.
// ---------------------------------------------------------------------------
__global__ void dirgat_bias(float* __restrict__ out,
                            const float* __restrict__ b1, const float* __restrict__ b2) {
  int t = blockIdx.x * blockDim.x + threadIdx.x;
  if (t >= NNODES * DOUT) return;
  int c = t & 63;
  out[t] += (1.0f - ALPHA_MIX) * b1[c] + ALPHA_MIX * b2[c];
}

// ---------------------------------------------------------------------------
extern "C" void kernel_launch(void* const* d_in, const int* in_sizes, int n_in,
                              void* d_out, int out_size, void* d_ws, size_t ws_size,
                              hipStream_t stream) {
  const float* x   = (const float*)d_in[0];
  const int*   ei  = (const int*)d_in[1];
  const float* W1  = (const float*)d_in[2];
  const float* as1 = (const float*)d_in[3];
  const float* ad1 = (const float*)d_in[4];
  const float* b1  = (const float*)d_in[5];
  const float* W2  = (const float*)d_in[6];
  const float* as2 = (const float*)d_in[7];
  const float* ad2 = (const float*)d_in[8];
  const float* b2  = (const float*)d_in[9];
  float* out = (float*)d_out;

  const int* srcp = ei;
  const int* dstp = ei + NEDGES;

  // Workspace carve-up (floats). Total ~8.5M floats (~34 MB).
  float* w    = (float*)d_ws;
  float* xw1  = w;
  float* xw2  = xw1 + (size_t)NNODES * DOUT;
  float* a_s1 = xw2 + (size_t)NNODES * DOUT;
  float* a_d1 = a_s1 + NNODES;
  float* a_s2 = a_d1 + NNODES;
  float* a_d2 = a_s2 + NNODES;
  float* m1   = a_d2 + NNODES;
  float* m2   = m1 + NNODES;
  float* den1 = m2 + NNODES;
  float* den2 = den1 + NNODES;
  float* ex1  = den2 + NNODES;
  float* ex2  = ex1 + (NEDGES + NNODES);

  const int ET = NEDGES + NNODES;

  dirgat_init<<<(NNODES * DOUT + 255) / 256, 256, 0, stream>>>(out, m1, m2, den1, den2);

  dim3 ggrid((ROWBLOCKS + 3) / 4, 2, 1);   // 4 waves/block, 2 directions
  dirgat_gemm_wmma<<<ggrid, 128, 0, stream>>>(x, W1, W2, xw1, xw2);

  dirgat_scores<<<(NNODES + 255) / 256, 256, 0, stream>>>(
      xw1, xw2, as1, ad1, as2, ad2, a_s1, a_d1, a_s2, a_d2);

  dirgat_edge_max<<<(ET + 255) / 256, 256, 0, stream>>>(
      srcp, dstp, a_s1, a_d1, a_s2, a_d2, m1, m2);

  dirgat_edge_exp<<<(ET + 255) / 256, 256, 0, stream>>>(
      srcp, dstp, a_s1, a_d1, a_s2, a_d2, m1, m2, den1, den2, ex1, ex2);

  dirgat_edge_coef<<<(ET + 255) / 256, 256, 0, stream>>>(
      srcp, dstp, den1, den2, ex1, ex2);

  long long scatter_threads = (long long)ET * 64;
  dirgat_edge_scatter<<<(int)((scatter_threads + 255) / 256), 256, 0, stream>>>(
      srcp, dstp, ex1, ex2, xw1, xw2, out);

  dirgat_bias<<<(NNODES * DOUT + 255) / 256, 256, 0, stream>>>(out, b1, b2);
}